// TemporalGCN_31258771980774
// MI455X (gfx1250) — compile-verified
//
#include <hip/hip_runtime.h>
#include <hip/hip_bf16.h>

// ---------------------------------------------------------------------------
// Two-layer GCN (PyG GCNConv semantics) for MI455X / gfx1250.
//
//   deg = 1 + scatter_add(1, dst);  dis = rsqrt(deg)
//   per layer: H = X@W ;  out = b + H*dis^2 + scatter_add(H[src]*dis[s]*dis[t], dst); relu
//   final = relu(layer2) + x
//
// Bound by edge scatter (~1.4 GB of gather+atomic traffic -> ~60us @ 23.3TB/s).
// GEMMs (1.6 GFLOP total) are done with fp32-exact V_WMMA_F32_16X16X4_F32 so we
// match the fp32 reference without precision loss while exercising the CDNA5
// matrix path. W (16 KB) is staged in LDS for the B fragments.
// ---------------------------------------------------------------------------

typedef __attribute__((ext_vector_type(2))) float v2f;
typedef __attribute__((ext_vector_type(8))) float v8f;

#define GCN_D 64

// ---- degree / normalization ------------------------------------------------

__global__ __launch_bounds__(256) void k_deg_init(float* deg, int n) {
    int i = blockIdx.x * 256 + threadIdx.x;
    if (i < n) deg[i] = 1.0f;                 // self-loop contributes 1
}

__global__ __launch_bounds__(256) void k_deg_scatter(const int* __restrict__ dst,
                                                     float* deg, int E) {
    int e = blockIdx.x * 256 + threadIdx.x;
    if (e < E) unsafeAtomicAdd(&deg[dst[e]], 1.0f);   // global_atomic_add_f32
}

__global__ __launch_bounds__(256) void k_rsqrt_inplace(float* dis, int n) {
    int i = blockIdx.x * 256 + threadIdx.x;
    if (i < n) dis[i] = 1.0f / sqrtf(dis[i]);  // deg >= 1 always (self-loops)
}

// ---- dense transform H = X @ W via V_WMMA_F32_16X16X4_F32 ------------------
// One wave per 16-row M-tile. 4 N-tile accumulators, K marched in steps of 4.
// A frag (16x4 f32, ISA layout): lane L<16 -> row m0+L, {K=k0,k0+1};
//                                lane L>=16 -> row m0+L-16, {K=k0+2,k0+3}.
// B frag (4x16 f32): lane L<16 -> col n0+L, {K=k0,k0+1}; upper half K=k0+2,+3.
// C/D (16x16 f32): VGPR r -> row m0+r (lanes 0-15) / m0+8+r (lanes 16-31).

__global__ __launch_bounds__(256) void k_gemm_wmma(const float* __restrict__ X,
                                                   const float* __restrict__ W,
                                                   float* __restrict__ H,
                                                   int nrows) {
    __shared__ float Wl[GCN_D * GCN_D];        // 16 KB of 320 KB/WGP LDS
    int t = threadIdx.x;
    // cooperative W load: 4096 floats = 1024 float4, 4 per thread
    {
        const float4* Wv  = (const float4*)W;
        float4*       Wlv = (float4*)Wl;
        for (int i = t; i < (GCN_D * GCN_D) / 4; i += 256) Wlv[i] = Wv[i];
    }
    __syncthreads();

    const int wave  = t >> 5;
    const int lane  = t & 31;
    const int mtile = blockIdx.x * 8 + wave;          // wave-uniform
    if (mtile * 16 >= nrows) return;                  // EXEC stays all-1s per wave
    const int m0   = mtile * 16;
    const int half = lane >> 4;                        // 0: K+0..1, 1: K+2..3
    const int lan  = lane & 15;
    const int row  = m0 + lan;
    const int klo  = half * 2;

    v8f acc0 = {}, acc1 = {}, acc2 = {}, acc3 = {};

    for (int k0 = 0; k0 < GCN_D; k0 += 4) {
        v2f a = *(const v2f*)(X + (size_t)row * GCN_D + k0 + klo);
        const float* wb = Wl + (k0 + klo) * GCN_D + lan;   // [K][N] row-major
        v2f b0, b1, b2, b3;
        b0.x = wb[0];       b0.y = wb[GCN_D];              // n0 = 0
        b1.x = wb[16];      b1.y = wb[GCN_D + 16];         // n0 = 16
        b2.x = wb[32];      b2.y = wb[GCN_D + 32];         // n0 = 32
        b3.x = wb[48];      b3.y = wb[GCN_D + 48];         // n0 = 48
        acc0 = __builtin_amdgcn_wmma_f32_16x16x4_f32(false, a, false, b0, (short)0, acc0, false, false);
        acc1 = __builtin_amdgcn_wmma_f32_16x16x4_f32(false, a, false, b1, (short)0, acc1, false, false);
        acc2 = __builtin_amdgcn_wmma_f32_16x16x4_f32(false, a, false, b2, (short)0, acc2, false, false);
        acc3 = __builtin_amdgcn_wmma_f32_16x16x4_f32(false, a, false, b3, (short)0, acc3, false, false);
    }

    // store: element r of acc -> row m0 + r + 8*half, col n0 + lan
    for (int r = 0; r < 8; ++r) {
        float* o = H + (size_t)(m0 + r + 8 * half) * GCN_D + lan;
        o[0]  = acc0[r];
        o[16] = acc1[r];
        o[32] = acc2[r];
        o[48] = acc3[r];
    }
}

// ---- aggregation -----------------------------------------------------------

// out[i,d] = bias[d] + H[i,d]*dis[i]^2   (bias + self-loop term; initializes out)
__global__ __launch_bounds__(256) void k_agg_init(const float* __restrict__ H,
                                                  const float* __restrict__ dis,
                                                  const float* __restrict__ bias,
                                                  float* __restrict__ out, int total) {
    int t = blockIdx.x * 256 + threadIdx.x;
    if (t < total) {
        int i = t >> 6, d = t & 63;
        float s = dis[i];
        out[t] = bias[d] + H[t] * s * s;
    }
}

// 16 lanes per edge, float4 per lane: coalesced 256B gather + 4 fp32 atomics
__global__ __launch_bounds__(256) void k_edge_scatter(const float* __restrict__ H,
                                                      const float* __restrict__ dis,
                                                      const int* __restrict__ src,
                                                      const int* __restrict__ dst,
                                                      float* __restrict__ out, int E) {
    int tid = blockIdx.x * 256 + threadIdx.x;
    int e   = tid >> 4;
    if (e >= E) return;
    int d4 = (tid & 15) * 4;
    int s = src[e], tdx = dst[e];
    float c = dis[s] * dis[tdx];
    const float4 v = *(const float4*)(H + (size_t)s * GCN_D + d4);
    float* o = out + (size_t)tdx * GCN_D + d4;
    unsafeAtomicAdd(o + 0, v.x * c);
    unsafeAtomicAdd(o + 1, v.y * c);
    unsafeAtomicAdd(o + 2, v.z * c);
    unsafeAtomicAdd(o + 3, v.w * c);
}

__global__ __launch_bounds__(256) void k_relu(float* buf, int total) {
    int t = blockIdx.x * 256 + threadIdx.x;
    if (t < total) buf[t] = fmaxf(buf[t], 0.0f);
}

// final = relu(out) + x
__global__ __launch_bounds__(256) void k_relu_residual(float* __restrict__ out,
                                                       const float* __restrict__ x,
                                                       int total) {
    int t = blockIdx.x * 256 + threadIdx.x;
    if (t < total) out[t] = fmaxf(out[t], 0.0f) + x[t];
}

// ---------------------------------------------------------------------------

static inline int gb(long long n) { return (int)((n + 255) / 256); }

extern "C" void kernel_launch(void* const* d_in, const int* in_sizes, int n_in,
                              void* d_out, int out_size, void* d_ws, size_t ws_size,
                              hipStream_t stream) {
    const float* x    = (const float*)d_in[0];
    const int*   edge = (const int*)d_in[1];
    const float* W1   = (const float*)d_in[2];
    const float* b1   = (const float*)d_in[3];
    const float* W2   = (const float*)d_in[4];
    const float* b2   = (const float*)d_in[5];

    const int N = in_sizes[0] / GCN_D;     // 100000
    const int E = in_sizes[1] / 2;         // 1250000
    const int* src = edge;                 // edge_index[0]
    const int* dst = edge + E;             // edge_index[1]
    const int total = N * GCN_D;

    char*  ws  = (char*)d_ws;
    float* dis = (float*)ws;                          // N floats (400 KB)
    float* A   = (float*)(ws + (size_t)(1  << 20));   // N*64 floats (25.6 MB)
    float* B   = (float*)(ws + (size_t)(28 << 20));   // N*64 floats (25.6 MB)
    float* out = (float*)d_out;

    const int mtiles      = (N + 15) / 16;
    const int gemm_blocks = (mtiles + 7) / 8;

    // normalization: deg -> dis = rsqrt(deg)
    k_deg_init   <<<gb(N), 256, 0, stream>>>(dis, N);
    k_deg_scatter<<<gb(E), 256, 0, stream>>>(dst, dis, E);
    k_rsqrt_inplace<<<gb(N), 256, 0, stream>>>(dis, N);

    // layer 1: A = x@W1 ; B = b1 + A*dis^2 + scatter(A) ; relu
    k_gemm_wmma   <<<gemm_blocks, 256, 0, stream>>>(x, W1, A, N);
    k_agg_init    <<<gb(total), 256, 0, stream>>>(A, dis, b1, B, total);
    k_edge_scatter<<<gb((long long)E * 16), 256, 0, stream>>>(A, dis, src, dst, B, E);
    k_relu        <<<gb(total), 256, 0, stream>>>(B, total);

    // layer 2: A = B@W2 ; out = b2 + A*dis^2 + scatter(A) ; relu + residual
    k_gemm_wmma   <<<gemm_blocks, 256, 0, stream>>>(B, W2, A, N);
    k_agg_init    <<<gb(total), 256, 0, stream>>>(A, dis, b2, out, total);
    k_edge_scatter<<<gb((long long)E * 16), 256, 0, stream>>>(A, dis, src, dst, out, E);
    k_relu_residual<<<gb(total), 256, 0, stream>>>(out, x, total);
}